// MoE_82454782149197
// MI455X (gfx1250) — compile-verified
//
#include <hip/hip_runtime.h>
#include <hip/hip_bf16.h>
#include <math.h>

// ---------------------------------------------------------------------------
// MoE forward for MI455X (gfx1250, wave32).
// Memory-bound (~236 MB HBM, multi-GB L2 weight streams, ~78 GF fp32):
//  - fp32 WMMA (V_WMMA_F32_16X16X4_F32) keeps full accuracy; matrix pipe is
//    nowhere near the bottleneck.
//  - 32-row M tiles (2 WMMA M-tiles) halve L2 weight traffic vs 16-row tiles.
//  - X-row gather uses GLOBAL_LOAD_ASYNC_TO_LDS_B128 (ASYNCcnt path).
// ---------------------------------------------------------------------------

typedef __attribute__((ext_vector_type(2))) float v2f;
typedef __attribute__((ext_vector_type(8))) float v8f;

#define TOKENS 4096      // B*S
#define DMODEL 1024      // D
#define NEXP   32        // E
#define TOPK   4
#define CAP    2048      // CAP_FACTOR * T * K / E
#define HROUT  512       // routed expert inner dim
#define HSHARE 1024      // shared expert inner dim (2*H)
#define XPAD   4         // LDS row padding (floats): stride%64==4 -> lanes 0-15
                         // hit banks {4m,4m+1}, lanes 16-31 {4m+2,4m+3}: no conflict
#define MROWS  32        // rows per workgroup tile (2 WMMA M-tiles)

// ---------------------------------------------------------------------------
// Kernel 0: reset counters & pre-fill dispatch lists (tok=0, w=0 => padded
// rows compute garbage * 0.0 and atomically add 0.0 -> harmless).
// ---------------------------------------------------------------------------
__global__ void moe_init(int* __restrict__ cnt, int* __restrict__ tokList,
                         float* __restrict__ wList) {
  int i = blockIdx.x * blockDim.x + threadIdx.x;
  if (i < NEXP) cnt[i] = 0;
  if (i < NEXP * CAP) {
    tokList[i] = 0;
    wList[i] = 0.0f;
  }
}

// ---------------------------------------------------------------------------
// Kernel 1: gate. One wave32 per token; lane == expert (E == 32 == wave32).
// ---------------------------------------------------------------------------
__global__ __launch_bounds__(256) void gate_kernel(
    const float* __restrict__ x, const float* __restrict__ Wg,
    int* __restrict__ cnt, int* __restrict__ tokList, float* __restrict__ wList) {
  const int lane = threadIdx.x & 31;
  const int wv = threadIdx.x >> 5;
  const int t = blockIdx.x * 8 + wv;
  if (t >= TOKENS) return;

  const float* xr = x + (size_t)t * DMODEL;
  float s = 0.0f;
  for (int d0 = 0; d0 < DMODEL; d0 += 32) {
    float xv = xr[d0 + lane];
#pragma unroll
    for (int j = 0; j < 32; ++j) {
      float xb = __shfl(xv, j, 32);
      s = fmaf(xb, Wg[(size_t)(d0 + j) * NEXP + lane], s);
    }
  }
  // softmax over the 32 experts (one per lane)
  float m = s;
#pragma unroll
  for (int off = 16; off > 0; off >>= 1) m = fmaxf(m, __shfl_xor(m, off, 32));
  float ex = __expf(s - m);
  float sum = ex;
#pragma unroll
  for (int off = 16; off > 0; off >>= 1) sum += __shfl_xor(sum, off, 32);
  const float prob = ex / sum;

  // iterative top-4 (lowest index wins ties, matching lax.top_k)
  float v = prob;
#pragma unroll
  for (int k = 0; k < TOPK; ++k) {
    float bv = v;
    int bi = lane;
#pragma unroll
    for (int off = 16; off > 0; off >>= 1) {
      float ov = __shfl_xor(bv, off, 32);
      int oi = __shfl_xor(bi, off, 32);
      if (ov > bv || (ov == bv && oi < bi)) { bv = ov; bi = oi; }
    }
    float bw = __shfl(prob, bi, 32);
    if (lane == k) {
      int pos = atomicAdd(&cnt[bi], 1);
      if (pos < CAP) {
        tokList[bi * CAP + pos] = t;
        wList[bi * CAP + pos] = bw;
      }
    }
    if (lane == bi) v = -3.0e38f;  // remove winner
  }
}

// ---------------------------------------------------------------------------
// Fused SwiGLU FFN tile kernel (routed + shared experts).
//   blockIdx.x = expert, blockIdx.y = 32-row tile.
//   GEMM1 (NC1 sequential N-chunks to cap live accumulators at 128 VGPRs):
//       X[32,D] @ W1/W3[D,HDIM] -> h = silu(.)*(.) into LDS
//   GEMM2: h[32,HDIM] @ W2[HDIM,D] -> weighted scatter-add (or plain store).
// WMMA operand mapping (ISA 7.12.2, f32 16x16x4):
//   A: lane m=lane&15, K pair {2*(lane>>4), +1}  -> ds_load_b64 from LDS
//   B: lane n=lane&15, same K pair               -> 2x global b32 (L2 stream),
//      each B fragment reused by 2 M-tiles (halves L2 weight traffic)
//   C/D: VGPR r -> M = r + 8*(lane>>4), N = lane&15
// ---------------------------------------------------------------------------
template <int HDIM, int NT1, int NC1, int NT2, bool ROUTED>
__global__ __launch_bounds__(256) void moe_ffn_kernel(
    const float* __restrict__ X, const float* __restrict__ W1,
    const float* __restrict__ W3, const float* __restrict__ W2,
    const int* __restrict__ cnt, const int* __restrict__ tokList,
    const float* __restrict__ wList, float* __restrict__ Y) {
  extern __shared__ float smem[];
  float* xs = smem;                               // [32][DMODEL+XPAD]
  float* hs = smem + MROWS * (DMODEL + XPAD);     // [32][HDIM+XPAD]
  __shared__ int s_tok[MROWS];
  __shared__ float s_w[MROWS];

  const int e = blockIdx.x;
  const int tile = blockIdx.y;
  const int tid = threadIdx.x;
  const int lane = tid & 31;
  const int wv = tid >> 5;           // 8 waves
  const int nlan = lane & 15;        // M-row for A / N-col for B,C,D
  const int khalf = lane >> 4;       // K-pair selector / C-row offset

  if (ROUTED) {
    if (tile * MROWS >= cnt[e]) return;  // block-uniform early exit
  }

  if (tid < MROWS) {
    if (ROUTED) {
      const int idx = e * CAP + tile * MROWS + tid;
      s_tok[tid] = tokList[idx];
      s_w[tid] = wList[idx];
    } else {
      s_tok[tid] = tile * MROWS + tid;
      s_w[tid] = 1.0f;
    }
  }
  __syncthreads();

  // --- async gather: 32 rows of X -> LDS via GLOBAL_LOAD_ASYNC_TO_LDS_B128.
  // GV mode: vdst = LDS byte offset (low 32 bits of generic LDS pointer),
  // vaddr = 64-bit global address. 16B alignment holds (row stride 4112B).
  for (int i = tid; i < MROWS * (DMODEL / 4); i += 256) {
    const int r = i / (DMODEL / 4);
    const int c4 = (i % (DMODEL / 4)) * 4;
    unsigned ldsoff = (unsigned)(uintptr_t)(xs + r * (DMODEL + XPAD) + c4);
    unsigned long long ga =
        (unsigned long long)(uintptr_t)(X + (size_t)s_tok[r] * DMODEL + c4);
    asm volatile("global_load_async_to_lds_b128 %0, %1, off"
                 :: "v"(ldsoff), "v"(ga) : "memory");
  }
  asm volatile("s_wait_asynccnt 0x0" ::: "memory");
  __syncthreads();

  const float* w1b = W1 + (size_t)e * DMODEL * HDIM;
  const float* w3b = W3 + (size_t)e * DMODEL * HDIM;
  const float* w2b = W2 + (size_t)e * HDIM * DMODEL;

  const v8f zero8 = {0.f, 0.f, 0.f, 0.f, 0.f, 0.f, 0.f, 0.f};

  // ---------------- GEMM1: acc1 = X@W1, acc3 = X@W3 ----------------
  for (int c1 = 0; c1 < NC1; ++c1) {
    v8f acc1[2][NT1], acc3[2][NT1];
#pragma unroll
    for (int mt = 0; mt < 2; ++mt)
#pragma unroll
      for (int i = 0; i < NT1; ++i) { acc1[mt][i] = zero8; acc3[mt][i] = zero8; }

    for (int kb = 0; kb < DMODEL; kb += 4) {
      const v2f a0 = *(const v2f*)(xs + nlan * (DMODEL + XPAD) + kb + 2 * khalf);
      const v2f a1 = *(const v2f*)(xs + (nlan + 16) * (DMODEL + XPAD) + kb + 2 * khalf);
#pragma unroll
      for (int nt = 0; nt < NT1; ++nt) {
        const int n = ((c1 * 8 + wv) * NT1 + nt) * 16 + nlan;
        const float* b1p = w1b + (size_t)(kb + 2 * khalf) * HDIM + n;
        const float* b3p = w3b + (size_t)(kb + 2 * khalf) * HDIM + n;
        v2f b1; b1[0] = b1p[0]; b1[1] = b1p[HDIM];
        v2f b3; b3[0] = b3p[0]; b3[1] = b3p[HDIM];
        acc1[0][nt] = __builtin_amdgcn_wmma_f32_16x16x4_f32(
            false, a0, false, b1, (short)0, acc1[0][nt], false, false);
        acc1[1][nt] = __builtin_amdgcn_wmma_f32_16x16x4_f32(
            false, a1, false, b1, (short)0, acc1[1][nt], false, false);
        acc3[0][nt] = __builtin_amdgcn_wmma_f32_16x16x4_f32(
            false, a0, false, b3, (short)0, acc3[0][nt], false, false);
        acc3[1][nt] = __builtin_amdgcn_wmma_f32_16x16x4_f32(
            false, a1, false, b3, (short)0, acc3[1][nt], false, false);
      }
    }

    // h = silu(acc1) * acc3 -> LDS (each wave owns disjoint columns)
#pragma unroll
    for (int mt = 0; mt < 2; ++mt) {
#pragma unroll
      for (int nt = 0; nt < NT1; ++nt) {
        const int nbase = ((c1 * 8 + wv) * NT1 + nt) * 16;
#pragma unroll
        for (int r = 0; r < 8; ++r) {
          const float a1v = acc1[mt][nt][r];
          const float a3v = acc3[mt][nt][r];
          const float hv = a1v * (1.0f / (1.0f + __expf(-a1v))) * a3v;
          const int mm = mt * 16 + r + 8 * khalf;
          hs[mm * (HDIM + XPAD) + nbase + nlan] = hv;
        }
      }
    }
  }
  __syncthreads();

  // ---------------- GEMM2: out = h @ W2 ----------------
  v8f acc2[2][NT2];
#pragma unroll
  for (int mt = 0; mt < 2; ++mt)
#pragma unroll
    for (int i = 0; i < NT2; ++i) acc2[mt][i] = zero8;

  for (int kb = 0; kb < HDIM; kb += 4) {
    const v2f a0 = *(const v2f*)(hs + nlan * (HDIM + XPAD) + kb + 2 * khalf);
    const v2f a1 = *(const v2f*)(hs + (nlan + 16) * (HDIM + XPAD) + kb + 2 * khalf);
#pragma unroll
    for (int nt = 0; nt < NT2; ++nt) {
      const int n = (wv * NT2 + nt) * 16 + nlan;
      const float* bp = w2b + (size_t)(kb + 2 * khalf) * DMODEL + n;
      v2f b; b[0] = bp[0]; b[1] = bp[DMODEL];
      acc2[0][nt] = __builtin_amdgcn_wmma_f32_16x16x4_f32(
          false, a0, false, b, (short)0, acc2[0][nt], false, false);
      acc2[1][nt] = __builtin_amdgcn_wmma_f32_16x16x4_f32(
          false, a1, false, b, (short)0, acc2[1][nt], false, false);
    }
  }

  // ---------------- combine: weighted scatter-add (or plain store) --------
#pragma unroll
  for (int mt = 0; mt < 2; ++mt) {
#pragma unroll
    for (int nt = 0; nt < NT2; ++nt) {
      const int nbase = (wv * NT2 + nt) * 16;
#pragma unroll
      for (int r = 0; r < 8; ++r) {
        const int mm = mt * 16 + r + 8 * khalf;
        const float val = acc2[mt][nt][r] * s_w[mm];
        const size_t off = (size_t)s_tok[mm] * DMODEL + nbase + nlan;
        if (ROUTED)
          atomicAdd(&Y[off], val);   // global_atomic_add_f32, no return
        else
          Y[off] = val;              // shared experts initialize Y (runs first)
      }
    }
  }
}

// ---------------------------------------------------------------------------
// Host-side launch
// ---------------------------------------------------------------------------
extern "C" void kernel_launch(void* const* d_in, const int* in_sizes, int n_in,
                              void* d_out, int out_size, void* d_ws, size_t ws_size,
                              hipStream_t stream) {
  // setup_inputs order: x, Wg, W1, W2, W3, Ws1, Ws2, Ws3  (note W2 before W3!)
  const float* x   = (const float*)d_in[0];
  const float* Wg  = (const float*)d_in[1];
  const float* W1  = (const float*)d_in[2];
  const float* W2  = (const float*)d_in[3];
  const float* W3  = (const float*)d_in[4];
  const float* Ws1 = (const float*)d_in[5];
  const float* Ws2 = (const float*)d_in[6];
  const float* Ws3 = (const float*)d_in[7];
  float* y = (float*)d_out;

  // workspace: [cnt: 32 ints pad 256B][tokList: E*CAP ints][wList: E*CAP floats]
  char* ws = (char*)d_ws;
  int* cnt = (int*)ws;
  int* tokList = (int*)(ws + 256);
  float* wList = (float*)(ws + 256 + (size_t)NEXP * CAP * sizeof(int));

  // 0) reset dispatch state (atomically-built lists must be rebuilt per call)
  moe_init<<<(NEXP * CAP + 255) / 256, 256, 0, stream>>>(cnt, tokList, wList);

  // 1) gating + top-4 dispatch (one wave per token, 8 tokens / block)
  gate_kernel<<<TOKENS / 8, 256, 0, stream>>>(x, Wg, cnt, tokList, wList);

  // 2) shared experts over all tokens: plain stores initialize y
  {
    const size_t shmem =
        (size_t)(MROWS * (DMODEL + XPAD) + MROWS * (HSHARE + XPAD)) * sizeof(float);
    // HDIM=1024: GEMM1 in 2 N-chunks of 4 tiles/wave; GEMM2 8 tiles/wave
    moe_ffn_kernel<HSHARE, 4, 2, DMODEL / 16 / 8, false>
        <<<dim3(1, TOKENS / MROWS), 256, shmem, stream>>>(
            x, Ws1, Ws3, Ws2, nullptr, nullptr, nullptr, y);
  }

  // 3) routed experts: capacity tiles, weighted atomic scatter-add into y
  {
    const size_t shmem =
        (size_t)(MROWS * (DMODEL + XPAD) + MROWS * (HROUT + XPAD)) * sizeof(float);
    // HDIM=512: GEMM1 single chunk of 4 tiles/wave; GEMM2 8 tiles/wave
    moe_ffn_kernel<HROUT, 4, 1, DMODEL / 16 / 8, true>
        <<<dim3(NEXP, CAP / MROWS), 256, shmem, stream>>>(
            x, W1, W3, W2, cnt, tokList, wList, y);
  }
}